// KANSpline1D_15410342658414
// MI455X (gfx1250) — compile-verified
//
#include <hip/hip_runtime.h>
#include <cstdint>

// KAN spline 1D: out = lerp(alpha-table) + beta*x + bias, per channel.
// Bandwidth-bound: ~67 MB traffic -> ~2.9 us floor @ 23.3 TB/s.
// CDNA5 paths: global_load_async_to_lds_b128 (4-buffer, depth-3 pipeline)
//              + s_wait_asynccnt immediates, LDS lerp-table gather.

#define BLOCK  256
#define NT     4              // tiles per block
#define BUFS   4              // LDS double^2 buffering
#define QPT    2              // float4 quads per thread per tile (32 B/lane)
#define TILE_Q (BLOCK * QPT)  // 512 float4 = 8 KB per tile

__device__ __forceinline__ void async_load_b128(unsigned lds_addr, const void* gaddr) {
  // VDST = LDS byte address (low 32 bits of generic shared ptr), VADDR = 64-bit global addr
  asm volatile("global_load_async_to_lds_b128 %0, %1, off"
               :: "v"(lds_addr), "v"(gaddr)
               : "memory");
}

template<int N>
__device__ __forceinline__ void wait_async() {
  asm volatile("s_wait_asynccnt %0" :: "n"(N) : "memory");
}

__global__ __launch_bounds__(BLOCK)
void kan_spline_kernel(const float* __restrict__ x,
                       const float* __restrict__ alpha,
                       const float* __restrict__ beta,
                       const float* __restrict__ bias,
                       const float* __restrict__ shift,
                       const float* __restrict__ scale,
                       float* __restrict__ out)
{
  __shared__ float  apad[24];          // [0,0,0,0, a0..a15, 0,0,0,0]
  __shared__ float2 apair[24];         // apair[j] = (apad[j], apad[j+1]) -> one ds_load_b64
  __shared__ float4 xbuf[BUFS][TILE_Q];// 32 KB async staging

  const int tid  = threadIdx.x;
  const int blk  = blockIdx.x;         // 1024 blocks: 2 per (b,c) slab
  const int bc   = blk >> 1;
  const int half = blk & 1;
  const int c    = bc & 63;

  if (tid < 24) {
    const int k = tid - 4;
    apad[tid] = (k >= 0 && k < 16) ? alpha[c * 16 + k] : 0.0f;
  }
  __syncthreads();
  if (tid < 23) apair[tid] = make_float2(apad[tid], apad[tid + 1]);
  __syncthreads();

  // wave-uniform per-channel params -> scalar loads
  const float sc  = scale[c];
  const float sh  = shift[c];
  const float be  = beta[c];
  const float bi  = bias[c];
  const float nss = -sh * sc;          // t = x*sc - sh*sc

  const size_t base = (size_t)bc * 16384u + (size_t)half * 8192u;
  const float4* gsrc = (const float4*)(x + base);
  float4*       gdst = (float4*)(out + base);

  unsigned lds_lo[BUFS], lds_hi[BUFS];
#pragma unroll
  for (int b = 0; b < BUFS; ++b) {
    lds_lo[b] = (unsigned)(uintptr_t)&xbuf[b][tid];
    lds_hi[b] = (unsigned)(uintptr_t)&xbuf[b][tid + BLOCK];
  }

  auto issue_tile = [&](int t) {
    const int b = t & (BUFS - 1);
    async_load_b128(lds_lo[b], gsrc + (size_t)t * TILE_Q + tid);
    async_load_b128(lds_hi[b], gsrc + (size_t)t * TILE_Q + BLOCK + tid);
  };

  auto lerp4 = [&](const float4 xv) -> float4 {
    const float in[4] = {xv.x, xv.y, xv.z, xv.w};
    float r[4];
#pragma unroll
    for (int i = 0; i < 4; ++i) {
      const float xi = in[i];
      float tt = __builtin_fmaf(xi, sc, nss);
      tt = fminf(1.5f, fmaxf(-1.5f, tt));
      const float pos = __builtin_fmaf(tt, 7.5f, 11.5f); // padded index in [0.25, 22.75]
      const float jf  = floorf(pos);
      const float w   = pos - jf;
      const float2 ap = apair[(int)jf];                  // ds_load_b64, 8B aligned
      const float phi = __builtin_fmaf(ap.y - ap.x, w, ap.x);
      r[i] = __builtin_fmaf(be, xi, bi) + phi;
    }
    return make_float4(r[0], r[1], r[2], r[3]);
  };

  auto compute_tile = [&](int t) {
    const int b = t & (BUFS - 1);
    const float4 v0 = xbuf[b][tid];            // ds_load_b128
    const float4 v1 = xbuf[b][tid + BLOCK];
    gdst[(size_t)t * TILE_Q + tid]         = lerp4(v0);  // global_store_b128
    gdst[(size_t)t * TILE_Q + BLOCK + tid] = lerp4(v1);
  };

  // Prologue: stage tiles 0..2 (depth-3 prefetch, 2 async ops per tile per wave).
  issue_tile(0);
  issue_tile(1);
  issue_tile(2);

  // Steady state, fully unrolled. Async loads complete in order, so before
  // consuming tile t we wait until outstanding <= 2 * (#tiles issued after t).
  issue_tile(3); wait_async<6>(); compute_tile(0);
                 wait_async<4>(); compute_tile(1);
                 wait_async<2>(); compute_tile(2);
                 wait_async<0>(); compute_tile(3);
}

extern "C" void kernel_launch(void* const* d_in, const int* in_sizes, int n_in,
                              void* d_out, int out_size, void* d_ws, size_t ws_size,
                              hipStream_t stream) {
  const float* x     = (const float*)d_in[0];
  const float* alpha = (const float*)d_in[1];
  const float* beta  = (const float*)d_in[2];
  const float* bias  = (const float*)d_in[3];
  const float* shift = (const float*)d_in[4];
  const float* scale = (const float*)d_in[5];
  float* out = (float*)d_out;

  dim3 grid(8 * 64 * 2);   // 2 blocks per (b,c) slab of 16384 elements
  dim3 block(BLOCK);
  hipLaunchKernelGGL(kan_spline_kernel, grid, block, 0, stream,
                     x, alpha, beta, bias, shift, scale, out);
}